// Matrix5ErrorJointFuzzifyLayer_75015898792515
// MI455X (gfx1250) — compile-verified
//
#include <hip/hip_runtime.h>

// 9-row fuzzify layer: y[k][i] = clip(diag[k] * aug_k(x[i]) + bias[k], 0, 1)
// Memory-bound streaming map: 32 MB in, 288 MB out -> ~13.7 us floor @ 23.3 TB/s.
// Strategy: b128 vector loads/stores, non-temporal hints, wave-uniform scalar math.

typedef __attribute__((ext_vector_type(4))) float v4f;

#define MIN_SLOPE 0.01f

__device__ __forceinline__ float satf(float v) {
    return fminf(fmaxf(v, 0.0f), 1.0f);
}

__global__ __launch_bounds__(256) void fuzzify9_kernel(
    const float* __restrict__ x,
    const float* __restrict__ g_td_c,
    const float* __restrict__ g_td_s,
    const float* __restrict__ g_dl_s,
    const float* __restrict__ g_dl_cw,
    const float* __restrict__ g_dl_sw,
    const float* __restrict__ g_dl_ssw,
    float* __restrict__ out,
    int n)
{
    // ---- wave-uniform parameter math (expect s_load_b32 + SGPR/low VALU) ----
    const float td_s    = fabsf(g_td_s[0] - MIN_SLOPE) + MIN_SLOPE;
    const float dl_s    = fabsf(g_dl_s[0] - MIN_SLOPE) + MIN_SLOPE;
    const float slope_w = 1.0f / dl_s;
    const float c_half  = 0.5f * fabsf(g_dl_cw[0]);
    const float sw      = fabsf(g_dl_sw[0]);
    const float ssw     = fabsf(g_dl_ssw[0]);
    const float s_half  = 0.5f * sw;
    const float ss_half = 0.5f * ssw;

    const float d1 = fmaf(dl_s, s_half, 1.0f);
    const float c1 = c_half + slope_w + s_half;
    const float d2 = fmaf(dl_s, ss_half, 1.0f);
    const float c2 = c_half + 2.0f * slope_w + sw + ss_half;
    const float c3 = c_half + sw + 3.0f * slope_w + ssw;
    const float d3 = fmaf(-dl_s, c3, 1.0f);
    const float a  = g_td_c[0] * td_s;

    const float diag[9] = {-td_s, td_s, dl_s, -dl_s, -dl_s, -dl_s, -dl_s, -dl_s, dl_s};
    const float bias[9] = {1.0f + a, -a, d3, d2, d1, fmaf(dl_s, c_half, 1.0f), d1, d2, d3};

    const long long i = 4LL * ((long long)blockIdx.x * blockDim.x + threadIdx.x);

    if (i + 3 < (long long)n) {
        // one global_load_b128 (non-temporal: input read exactly once)
        v4f xv = __builtin_nontemporal_load((const v4f*)(x + i));

        v4f aug[9];
        aug[0] = xv;
        aug[1] = xv;
        aug[8] = xv;
        #pragma unroll
        for (int j = 0; j < 4; ++j) {
            aug[2][j] = -xv[j];
            aug[3][j] = fabsf(xv[j] + c2);
            aug[4][j] = fabsf(xv[j] + c1);
            aug[5][j] = fabsf(xv[j]);
            aug[6][j] = fabsf(xv[j] - c1);
            aug[7][j] = fabsf(xv[j] - c2);
        }

        // nine global_store_b128, TH=NT (288 MB stream > 192 MB L2, never re-read)
        #pragma unroll
        for (int k = 0; k < 9; ++k) {
            v4f y;
            #pragma unroll
            for (int j = 0; j < 4; ++j)
                y[j] = satf(fmaf(diag[k], aug[k][j], bias[k]));
            __builtin_nontemporal_store(y, (v4f*)(out + (long long)k * n + i));
        }
    } else if (i < (long long)n) {
        // scalar tail (unused for N = 8,388,608, kept for generality)
        for (long long e = i; e < (long long)n; ++e) {
            const float xe = x[e];
            const float aug_s[9] = {
                xe, xe, -xe,
                fabsf(xe + c2), fabsf(xe + c1), fabsf(xe),
                fabsf(xe - c1), fabsf(xe - c2), xe
            };
            for (int k = 0; k < 9; ++k)
                out[(long long)k * n + e] = satf(fmaf(diag[k], aug_s[k], bias[k]));
        }
    }
}

extern "C" void kernel_launch(void* const* d_in, const int* in_sizes, int n_in,
                              void* d_out, int out_size, void* d_ws, size_t ws_size,
                              hipStream_t stream) {
    const float* x      = (const float*)d_in[0];
    const float* td_c   = (const float*)d_in[1];
    const float* td_s   = (const float*)d_in[2];
    const float* dl_s   = (const float*)d_in[3];
    const float* dl_cw  = (const float*)d_in[4];
    const float* dl_sw  = (const float*)d_in[5];
    const float* dl_ssw = (const float*)d_in[6];
    float* out = (float*)d_out;

    const int n = in_sizes[0];
    const long long nThreads = ((long long)n + 3) / 4;
    const int block = 256;  // 8 wave32 waves per workgroup
    const unsigned grid = (unsigned)((nThreads + block - 1) / block);

    fuzzify9_kernel<<<dim3(grid), dim3(block), 0, stream>>>(
        x, td_c, td_s, dl_s, dl_cw, dl_sw, dl_ssw, out, n);
}